// LF_Net_39006892982699
// MI455X (gfx1250) — compile-verified
//
#include <hip/hip_runtime.h>
#include <hip/hip_bf16.h>
#include <math.h>

// CDNA5 (gfx1250) wave32 WMMA, full-precision f32 path: V_WMMA_F32_16X16X4_F32.
typedef float v2f __attribute__((ext_vector_type(2)));
typedef float v8f __attribute__((ext_vector_type(8)));

__device__ __forceinline__ v8f wmma4(v2f a, v2f b, v8f c) {
  // (neg_a, A, neg_b, B, c_mod, C, reuse_a, reuse_b)
  return __builtin_amdgcn_wmma_f32_16x16x4_f32(false, a, false, b, (short)0, c,
                                               false, false);
}

__device__ __forceinline__ v2f ld2(const float* p) { return *(const v2f*)p; }

// ---------------------------------------------------------------------------
// Workgroup-tiled GEMM, 256 threads = 8 wave32, block tile 128(M) x 64(N).
// All 8 waves share the same 64 output columns -> the 64xKC B tile is staged
// cooperatively into LDS, double buffered, in a bank-conflict-free layout
// ([col][k], row stride 68 floats; 68 == 4 mod 64 banks).
//
// BT  = true : Bsrc is W[N,K] row-major (C = A @ W^T), staging is a plain copy
// BT  = false: Bsrc is B[K,N] row-major (C = A @ B),  staging transposes
// EPI = 0: C = tanh(acc + aux[col])      (aux = bias[N])
// EPI = 1: C = acc * (1 - aux[row,col]^2) (aux = a1, derivative mask, MxN)
// EPI = 2: C = acc
// ---------------------------------------------------------------------------
constexpr int KC  = 64;       // K-chunk staged per double-buffer step
constexpr int LDB = KC + 4;   // padded LDS row stride (floats): 68 % 64 == 4

template <int EPI, bool BT>
__global__ __launch_bounds__(256)
void gemm_wmma(const float* __restrict__ A, const float* __restrict__ Bsrc,
               const float* __restrict__ aux, float* __restrict__ C,
               int N, int K) {
  __shared__ float lb[2][64 * LDB];

  const int tid    = threadIdx.x;
  const int lane   = tid & 31;
  const int lrow   = lane & 15;   // M index (A frag) / N index (B frag)
  const int khalf  = lane >> 4;   // which K pair this lane holds
  const int widx   = tid >> 5;    // wave in block: 0..7
  const int ntiles = N >> 6;
  const int nbase  = (blockIdx.x % ntiles) << 6;
  const int mbase  = ((blockIdx.x / ntiles) << 7) + (widx << 4);

  auto stage = [&](int buf, int k0) {
    if constexpr (BT) {
      // lb[c][kk] = W[nbase + c][k0 + kk] : contiguous float4 copy
      for (int i = tid; i < 64 * (KC / 4); i += 256) {
        const int c  = i / (KC / 4);
        const int kk = (i % (KC / 4)) * 4;
        const float4 v =
            *(const float4*)(Bsrc + (size_t)(nbase + c) * K + k0 + kk);
        *(float4*)(&lb[buf][c * LDB + kk]) = v;
      }
    } else {
      // lb[c][kk] = B[k0 + kk][nbase + c] : transpose, coalesced over c
      for (int i = tid; i < 64 * KC; i += 256) {
        const int c  = i & 63;
        const int kk = i >> 6;
        lb[buf][c * LDB + kk] = Bsrc[(size_t)(k0 + kk) * N + nbase + c];
      }
    }
  };

  v8f acc[4] = {};
  const float* arow = A + (size_t)(mbase + lrow) * K + khalf * 2;

  stage(0, 0);
  __syncthreads();
  int buf = 0;
  for (int k0 = 0; k0 < K; k0 += KC) {
    if (k0 + KC < K) {
      __builtin_prefetch(arow + k0 + KC, 0, 1);  // global_prefetch for next A
      stage(buf ^ 1, k0 + KC);                   // overlaps with WMMA below
    }
#pragma unroll
    for (int kk = 0; kk < KC; kk += 4) {
      const v2f a  = ld2(arow + k0 + kk);
      const v2f b0 = ld2(&lb[buf][( 0 + lrow) * LDB + kk + khalf * 2]);
      const v2f b1 = ld2(&lb[buf][(16 + lrow) * LDB + kk + khalf * 2]);
      const v2f b2 = ld2(&lb[buf][(32 + lrow) * LDB + kk + khalf * 2]);
      const v2f b3 = ld2(&lb[buf][(48 + lrow) * LDB + kk + khalf * 2]);
      acc[0] = wmma4(a, b0, acc[0]);
      acc[1] = wmma4(a, b1, acc[1]);
      acc[2] = wmma4(a, b2, acc[2]);
      acc[3] = wmma4(a, b3, acc[3]);
    }
    __syncthreads();  // staging of buf^1 done; everyone finished reading buf
    buf ^= 1;
  }

#pragma unroll
  for (int j = 0; j < 4; ++j) {
    const int col = nbase + j * 16 + lrow;
    float bv = 0.0f;
    if constexpr (EPI == 0) bv = aux[col];
#pragma unroll
    for (int r = 0; r < 8; ++r) {
      const int row = mbase + khalf * 8 + r;
      float v = acc[j][r];
      if constexpr (EPI == 0) v = tanhf(v + bv);
      if constexpr (EPI == 1) {
        const float m = aux[(size_t)row * N + col];
        v *= (1.0f - m * m);
      }
      C[(size_t)row * N + col] = v;
    }
  }
}

// ---------------------------------------------------------------------------
// Elementwise from a2 (2048x512): V[b] = 0.5*sum(a2^2), g2 = a2*(1-a2^2)
// (g2 may alias a2: pure per-element in-place rewrite)
// ---------------------------------------------------------------------------
__global__ __launch_bounds__(256)
void vg2_kernel(const float* __restrict__ a2, float* __restrict__ g2,
                float* __restrict__ Vout) {
  const int b = blockIdx.x;
  const int tid = threadIdx.x;
  float sum = 0.0f;
  for (int h = tid; h < 512; h += 256) {
    const float v = a2[(size_t)b * 512 + h];
    sum += v * v;
    g2[(size_t)b * 512 + h] = v * (1.0f - v * v);
  }
  __shared__ float red[256];
  red[tid] = sum;
  __syncthreads();
  for (int s = 128; s > 0; s >>= 1) {
    if (tid < s) red[tid] += red[tid + s];
    __syncthreads();
  }
  if (tid == 0) Vout[b] = 0.5f * red[0];
}

// ---------------------------------------------------------------------------
// u[M,16] = A[M,K] @ W[16,K]^T + bias[16].  W (16x512 = 32KB) is staged whole
// into LDS once (stride 516 floats, conflict-free), shared by all 8 waves.
// ---------------------------------------------------------------------------
__global__ __launch_bounds__(256)
void gemm_bt_bias16(const float* __restrict__ A, const float* __restrict__ W,
                    const float* __restrict__ bias, float* __restrict__ C,
                    int K) {
  constexpr int LDW = 512 + 4;
  __shared__ float lw[16 * LDW];

  const int tid   = threadIdx.x;
  const int lane  = tid & 31;
  const int lrow  = lane & 15;
  const int khalf = lane >> 4;
  const int wave  = blockIdx.x * (blockDim.x >> 5) + (tid >> 5);
  const int mbase = wave << 4;

  for (int i = tid; i < 16 * (512 / 4); i += 256) {
    const int c  = i / 128;
    const int kk = (i % 128) * 4;
    const float4 v = *(const float4*)(W + (size_t)c * 512 + kk);
    *(float4*)(&lw[c * LDW + kk]) = v;
  }
  __syncthreads();

  v8f acc = {};
  const float* arow = A + (size_t)(mbase + lrow) * K + khalf * 2;
#pragma unroll 8
  for (int k = 0; k < K; k += 4) {
    const v2f a = ld2(arow + k);
    const v2f w = ld2(&lw[lrow * LDW + k + khalf * 2]);
    acc = wmma4(a, w, acc);
  }

  const float bv = bias[lrow];
#pragma unroll
  for (int r = 0; r < 8; ++r)
    C[(size_t)(mbase + khalf * 8 + r) * 16 + lrow] = acc[r] + bv;
}

// ---------------------------------------------------------------------------
// f = x @ dynA^T + u @ dynB^T  (16x64 strip), then Vdot[b] = dot(gradV[b], f[b])
// reduced across the 16 lanes of each half-wave with shfl_xor.
// ---------------------------------------------------------------------------
__global__ __launch_bounds__(256)
void f_vdot_kernel(const float* __restrict__ x, const float* __restrict__ dynA,
                   const float* __restrict__ u, const float* __restrict__ dynB,
                   const float* __restrict__ gradV, float* __restrict__ vdot) {
  const int lane  = threadIdx.x & 31;
  const int lrow  = lane & 15;
  const int khalf = lane >> 4;
  const int wave  = blockIdx.x * (blockDim.x >> 5) + (threadIdx.x >> 5);
  const int mbase = wave << 4;

  v8f acc[4] = {};
  const float* arow = x + (size_t)(mbase + lrow) * 64 + khalf * 2;
#pragma unroll 4
  for (int k = 0; k < 64; k += 4) {
    const v2f a  = ld2(arow + k);
    const v2f b0 = ld2(dynA + (size_t)( 0 + lrow) * 64 + k + khalf * 2);
    const v2f b1 = ld2(dynA + (size_t)(16 + lrow) * 64 + k + khalf * 2);
    const v2f b2 = ld2(dynA + (size_t)(32 + lrow) * 64 + k + khalf * 2);
    const v2f b3 = ld2(dynA + (size_t)(48 + lrow) * 64 + k + khalf * 2);
    acc[0] = wmma4(a, b0, acc[0]);
    acc[1] = wmma4(a, b1, acc[1]);
    acc[2] = wmma4(a, b2, acc[2]);
    acc[3] = wmma4(a, b3, acc[3]);
  }
  const float* urow = u + (size_t)(mbase + lrow) * 16 + khalf * 2;
#pragma unroll
  for (int k = 0; k < 16; k += 4) {
    const v2f a  = ld2(urow + k);
    const v2f b0 = ld2(dynB + (size_t)( 0 + lrow) * 16 + k + khalf * 2);
    const v2f b1 = ld2(dynB + (size_t)(16 + lrow) * 16 + k + khalf * 2);
    const v2f b2 = ld2(dynB + (size_t)(32 + lrow) * 16 + k + khalf * 2);
    const v2f b3 = ld2(dynB + (size_t)(48 + lrow) * 16 + k + khalf * 2);
    acc[0] = wmma4(a, b0, acc[0]);
    acc[1] = wmma4(a, b1, acc[1]);
    acc[2] = wmma4(a, b2, acc[2]);
    acc[3] = wmma4(a, b3, acc[3]);
  }
#pragma unroll
  for (int r = 0; r < 8; ++r) {
    const int row = mbase + khalf * 8 + r;
    float p = 0.0f;
#pragma unroll
    for (int j = 0; j < 4; ++j)
      p += gradV[(size_t)row * 64 + j * 16 + lrow] * acc[j][r];
    p += __shfl_xor(p, 1, 32);
    p += __shfl_xor(p, 2, 32);
    p += __shfl_xor(p, 4, 32);
    p += __shfl_xor(p, 8, 32);
    if (lrow == 0) vdot[row] = p;
  }
}

// ---------------------------------------------------------------------------
extern "C" void kernel_launch(void* const* d_in, const int* in_sizes, int n_in,
                              void* d_out, int out_size, void* d_ws, size_t ws_size,
                              hipStream_t stream) {
  (void)in_sizes; (void)n_in; (void)out_size; (void)ws_size;

  const float* x     = (const float*)d_in[0];
  const float* V_W1  = (const float*)d_in[1];
  const float* V_b1  = (const float*)d_in[2];
  const float* V_W2  = (const float*)d_in[3];
  const float* V_b2  = (const float*)d_in[4];
  const float* U_W1  = (const float*)d_in[5];
  const float* U_b1  = (const float*)d_in[6];
  const float* U_W2  = (const float*)d_in[7];
  const float* U_b2  = (const float*)d_in[8];
  const float* U_W3  = (const float*)d_in[9];
  const float* U_b3  = (const float*)d_in[10];
  const float* dyn_A = (const float*)d_in[11];
  const float* dyn_B = (const float*)d_in[12];

  constexpr int BATCH = 2048, NH = 512, NI = 64;

  float* out    = (float*)d_out;
  float* out_u  = out;                 // (2048,16)
  float* out_V  = out + BATCH * 16;    // (2048,)
  float* out_Vd = out_V + BATCH;       // (2048,)

  float* ws = (float*)d_ws;
  const size_t P = (size_t)BATCH * NH;       // 1M floats per 2048x512 buffer
  float* a1    = ws;                         // a1 (kept for derivative mask)
  float* u1s   = ws + P;                     // u1, later reused for s
  float* a2g   = ws + 2 * P;                 // a2, rewritten in-place to g2
  float* u2    = ws + 3 * P;                 // u2
  float* gradV = ws + 4 * P;                 // (2048,64)

  const dim3 blk(256);                             // 8 wave32 per block
  const dim3 gBig((BATCH / 128) * (NH / 64));      // 128 blocks (128x64 tiles)
  const dim3 gGrad((BATCH / 128) * (NI / 64));     // 16 blocks
  const dim3 gRow((BATCH / 16) / 8);               // 16 blocks

  // Forward hidden layers (fused bias+tanh epilogue); C = A @ W^T
  gemm_wmma<0, true><<<gBig, blk, 0, stream>>>(x,   V_W1, V_b1, a1,  NH, NI);
  gemm_wmma<0, true><<<gBig, blk, 0, stream>>>(x,   U_W1, U_b1, u1s, NH, NI);
  gemm_wmma<0, true><<<gBig, blk, 0, stream>>>(a1,  V_W2, V_b2, a2g, NH, NH);
  gemm_wmma<0, true><<<gBig, blk, 0, stream>>>(u1s, U_W2, U_b2, u2,  NH, NH);

  // V = 0.5*sum(a2^2); g2 = a2*(1-a2^2)  (in-place over a2)
  vg2_kernel<<<dim3(BATCH), blk, 0, stream>>>(a2g, a2g, out_V);

  // Collapsed VJP: s = (g2 @ V_W2) * (1-a1^2); grad_V = s @ V_W1   (C = A @ B)
  gemm_wmma<1, false><<<gBig,  blk, 0, stream>>>(a2g, V_W2, a1,      u1s,   NH, NH);
  gemm_wmma<2, false><<<gGrad, blk, 0, stream>>>(u1s, V_W1, nullptr, gradV, NI, NH);

  // u = u2 @ U_W3^T + U_b3  -> output
  gemm_bt_bias16<<<gRow, blk, 0, stream>>>(u2, U_W3, U_b3, out_u, NH);

  // f = x @ dyn_A^T + u @ dyn_B^T;  Vdot = <grad_V, f>  -> output
  f_vdot_kernel<<<gRow, blk, 0, stream>>>(x, dyn_A, out_u, dyn_B, gradV, out_Vd);
}